// SpatioTemporal_Convolution_63711544868882
// MI455X (gfx1250) — compile-verified
//
#include <hip/hip_runtime.h>

typedef __attribute__((ext_vector_type(2))) float v2f;
typedef __attribute__((ext_vector_type(4))) float v4f;
typedef __attribute__((ext_vector_type(8))) float v8f;

#define NN    2048
#define BB    8
#define TT    24
#define CIN   32
#define COUT  64
#define KORD  3
#define JTOT  (TT*CIN)      // 768 = GEMM N-dim for rhs
#define MT    256           // workgroup M tile
#define NTL   64            // workgroup N tile
#define KC    16            // K slab per LDS stage
#define ASTR  20            // A tile LDS row stride (floats): even, 16B rows, bank-clean
#define BSTR  160           // B tile pair-row stride (floats): k-pair interleaved layout
#define A_FL  (MT*ASTR)     // 5120 floats
#define B_FL  (8*BSTR)      // 1280 floats
#define BUF_FL (2*A_FL + B_FL)  // 11520 floats per buffer

// ---- CDNA5 async global->LDS copies (ASYNCcnt) ----
__device__ __forceinline__ void async_ld128(unsigned lds_addr, unsigned goff,
                                            const float* sbase) {
    asm volatile("global_load_async_to_lds_b128 %0, %1, %2"
                 :: "v"(lds_addr), "v"(goff), "s"(sbase) : "memory");
}
__device__ __forceinline__ void async_ld32(unsigned lds_addr, unsigned goff,
                                           const float* sbase) {
    asm volatile("global_load_async_to_lds_b32 %0, %1, %2"
                 :: "v"(lds_addr), "v"(goff), "s"(sbase) : "memory");
}
__device__ __forceinline__ void wait_async0() {
    asm volatile("s_wait_asynccnt 0x0" ::: "memory");
}

// ---------------- power iteration (lambda_max) ----------------

__global__ void k_fill(float* v, float val, int n) {
    int i = blockIdx.x * 256 + threadIdx.x;
    if (i < n) v[i] = val;
}

__global__ __launch_bounds__(256) void k_matvec(const float* __restrict__ A,
                                                const float* __restrict__ v,
                                                float* __restrict__ y) {
    int row  = blockIdx.x * 8 + (threadIdx.x >> 5);   // one wave32 per row
    int lane = threadIdx.x & 31;
    const float* a = A + (size_t)row * NN;
    float s = 0.f;
    for (int j = lane; j < NN; j += 32) s += a[j] * v[j];
    #pragma unroll
    for (int off = 16; off; off >>= 1) s += __shfl_xor(s, off, 32);
    if (lane == 0) y[row] = s;
}

// mode 0: scal[0] = rsqrt(sum y^2) ; mode 1: scal[0] = dot(y, v)  (= lambda)
__global__ __launch_bounds__(256) void k_reduce(const float* __restrict__ y,
                                                const float* __restrict__ v,
                                                float* scal, int mode) {
    __shared__ float sm[256];
    float s = 0.f;
    for (int i = threadIdx.x; i < NN; i += 256)
        s += (mode == 0) ? y[i] * y[i] : y[i] * v[i];
    sm[threadIdx.x] = s;
    __syncthreads();
    for (int st = 128; st; st >>= 1) {
        if (threadIdx.x < st) sm[threadIdx.x] += sm[threadIdx.x + st];
        __syncthreads();
    }
    if (threadIdx.x == 0)
        scal[0] = (mode == 0) ? rsqrtf(fmaxf(sm[0], 1e-30f)) : sm[0];
}

__global__ void k_scale(const float* __restrict__ y, const float* __restrict__ scal,
                        float* __restrict__ v) {
    int i = blockIdx.x * 256 + threadIdx.x;
    if (i < NN) v[i] = y[i] * scal[0];
}

// ---------------- L_tilde = 2*adj/lam - I ----------------

__global__ __launch_bounds__(256) void k_ltilde(const float* __restrict__ adj,
                                                const float* __restrict__ scal,
                                                float* __restrict__ L) {
    size_t i = (size_t)blockIdx.x * 256 + threadIdx.x;   // 4M elements
    int n = (int)(i / NN), m = (int)(i % NN);
    L[i] = 2.f * adj[i] / scal[0] - (n == m ? 1.f : 0.f);
}

// ---------------- x transpose: xt[b][m][t][c] = x[b][t][m][c] ----------------

__global__ __launch_bounds__(256) void k_xt(const float* __restrict__ x,
                                            float* __restrict__ xt) {
    size_t i = (size_t)blockIdx.x * 256 + threadIdx.x;   // B*N*T*C, c fastest
    int c = (int)(i & (CIN - 1));
    size_t r = i >> 5;
    int t = (int)(r % TT); r /= TT;
    int m = (int)(r % NN);
    int b = (int)(r / NN);
    xt[i] = x[(((size_t)(b * TT + t) * NN) + m) * CIN + c];
}

// ---------------- WMMA f32 GEMM, async double-buffered ----------------
// C[M x Nj] = alpha * (A (∘ Att)) [M x 2048] @ B [2048 x Nj]  (- I if subI)
template <bool MUL_ATT>
__global__ __launch_bounds__(256) void k_gemm(const float* __restrict__ A1,
                                              const float* __restrict__ Att,
                                              const float* __restrict__ Bm,
                                              float* __restrict__ Cm,
                                              int Nj, float alpha, int subI) {
    __shared__ float smem[2 * BUF_FL];
    const unsigned lds0 = (unsigned)(unsigned long long)(uintptr_t)&smem[0];

    const int m0   = blockIdx.x * MT;
    const int j0   = blockIdx.y * NTL;
    const int tid  = threadIdx.x;
    const int wave = tid >> 5;
    const int lane = tid & 31;
    const int lh   = lane >> 4;      // half-wave: K offset 0/2, M offset 0/8
    const int l16  = lane & 15;
    const int wm   = wave * 32;      // wave's 32-row block inside tile

    v8f acc[2][4];
    #pragma unroll
    for (int ms = 0; ms < 2; ++ms)
        #pragma unroll
        for (int jt = 0; jt < 4; ++jt)
            #pragma unroll
            for (int e = 0; e < 8; ++e) acc[ms][jt][e] = 0.f;

    // issue async copies for K-slab kk0 into buffer buf
    auto stage = [&](int kk0, int buf) {
        unsigned lb = lds0 + (unsigned)(buf * (BUF_FL * 4));
        // A (and Att) tiles: 256x16 floats = 1024 b128 chunks; 4 per thread
        #pragma unroll
        for (int p = 0; p < 4; ++p) {
            int chunk = tid + p * 256;
            int row = chunk >> 2;
            int c4  = (chunk & 3) * 4;
            unsigned goff = (unsigned)(((m0 + row) * NN + kk0 + c4) * 4);
            unsigned la   = lb + (unsigned)((row * ASTR + c4) * 4);
            async_ld128(la, goff, A1);
            if (MUL_ATT) async_ld128(la + A_FL * 4, goff, Att);
        }
        // B tile: 16x64 floats, k-pair interleaved: lB[(k>>1)*BSTR + n*2 + (k&1)]
        #pragma unroll
        for (int p = 0; p < 4; ++p) {
            int f = tid + p * 256;
            int k = f >> 6;          // 0..15
            int n = f & 63;
            unsigned goff = (unsigned)(((kk0 + k) * Nj + j0 + n) * 4);
            unsigned la   = lb + (unsigned)((2 * A_FL + (k >> 1) * BSTR + n * 2 + (k & 1)) * 4);
            async_ld32(la, goff, Bm);
        }
    };

    stage(0, 0);
    const int NITER = NN / KC;   // 128
    for (int kt = 0; kt < NITER; ++kt) {
        int buf = kt & 1;
        wait_async0();           // my copies into buf are done
        __syncthreads();         // everyone's copies done; prev reads of buf^1 done
        if (kt + 1 < NITER) stage((kt + 1) * KC, buf ^ 1);

        const float* sA = smem + buf * BUF_FL;
        const float* sW = sA + A_FL;
        const float* sB = sA + 2 * A_FL;

        #pragma unroll
        for (int kk = 0; kk < KC; kk += 4) {
            // A frags (16x4 f32): lanes 0-15 K={0,1}, lanes 16-31 K={2,3}; one b64 each
            v2f af[2];
            #pragma unroll
            for (int ms = 0; ms < 2; ++ms) {
                int r = wm + ms * 16 + l16;
                v2f a = *(const v2f*)(sA + r * ASTR + kk + lh * 2);
                if (MUL_ATT) {
                    v2f w = *(const v2f*)(sW + r * ASTR + kk + lh * 2);
                    a *= w;                 // fused cheb ∘ att at fragment read
                }
                af[ms] = a;
            }
            #pragma unroll
            for (int jt = 0; jt < 4; ++jt) {
                // B frag (4x16 f32): pair row (kk/2 + lh), one aligned b64 load
                v2f bf = *(const v2f*)(sB + ((kk >> 1) + lh) * BSTR + (jt * 16 + l16) * 2);
                acc[0][jt] = __builtin_amdgcn_wmma_f32_16x16x4_f32(
                    false, af[0], false, bf, (short)0, acc[0][jt], false, false);
                acc[1][jt] = __builtin_amdgcn_wmma_f32_16x16x4_f32(
                    false, af[1], false, bf, (short)0, acc[1][jt], false, false);
            }
        }
        // no trailing barrier: next iteration's wait+barrier fences buf reuse
    }

    // epilogue: C/D VGPR v holds row v (lanes 0-15) and v+8 (lanes 16-31)
    #pragma unroll
    for (int ms = 0; ms < 2; ++ms)
        #pragma unroll
        for (int jt = 0; jt < 4; ++jt)
            #pragma unroll
            for (int v = 0; v < 8; ++v) {
                int row = m0 + wm + ms * 16 + v + lh * 8;
                int col = j0 + jt * 16 + l16;
                float r = alpha * acc[ms][jt][v];
                if (subI && row == col) r -= 1.f;
                Cm[(size_t)row * Nj + col] = r;
            }
}

// ---------------- k=0 term: rhs0[n][t][c] = att[n][n] * xt[n][t][c] ----------------

__global__ __launch_bounds__(256) void k_rhs0(const float* __restrict__ att_b,
                                              const float* __restrict__ xt_b,
                                              float* __restrict__ rhs0) {
    size_t i = (size_t)blockIdx.x * 256 + threadIdx.x;   // N*T*C
    size_t n = i / JTOT;
    rhs0[i] = att_b[n * NN + n] * xt_b[i];
}

// ---------------- theta contraction + ReLU: g[t][n][o] ----------------

__global__ __launch_bounds__(256) void k_theta(const float* __restrict__ rhs,
                                               const float* __restrict__ theta,
                                               float* __restrict__ g) {
    __shared__ float th[KORD * CIN * COUT];   // 6144 floats
    for (int i = threadIdx.x; i < KORD * CIN * COUT; i += 256) th[i] = theta[i];
    __syncthreads();
    int o  = threadIdx.x & 63;
    int ln = threadIdx.x >> 6;
    int tn = blockIdx.x * 4 + ln;             // flat over t*NN + n
    int t  = tn / NN;
    int n  = tn % NN;
    float s = 0.f;
    #pragma unroll
    for (int k = 0; k < KORD; ++k) {
        const float* rp = rhs + ((size_t)k * NN + n) * JTOT + t * CIN;
        const float* tp = th + k * CIN * COUT + o;
        #pragma unroll
        for (int c = 0; c < CIN; ++c) s += rp[c] * tp[c * COUT];
    }
    g[(size_t)tn * COUT + o] = fmaxf(s, 0.f);
}

// ---------------- temporal conv (1,3) + bias + ReLU ----------------

__global__ __launch_bounds__(256) void k_conv(const float* __restrict__ g,
                                              const float* __restrict__ w,
                                              const float* __restrict__ bias,
                                              float* __restrict__ out) {
    __shared__ float wt[3 * COUT * COUT];     // wt[dt][oi][o]: lane-contiguous in o
    __shared__ float bsh[COUT];
    for (int i = threadIdx.x; i < 3 * COUT * COUT; i += 256) {
        int o = i / (COUT * 3);
        int r = i % (COUT * 3);
        int oi = r / 3, dt = r % 3;
        wt[(dt * COUT + oi) * COUT + o] = w[i];
    }
    if (threadIdx.x < COUT) bsh[threadIdx.x] = bias[threadIdx.x];
    __syncthreads();
    int o  = threadIdx.x & 63;
    int ln = threadIdx.x >> 6;
    int tn = blockIdx.x * 4 + ln;
    int t  = tn / NN;
    int n  = tn % NN;
    float s = bsh[o];
    #pragma unroll
    for (int dt = -1; dt <= 1; ++dt) {
        int t2 = t + dt;
        if (t2 < 0 || t2 >= TT) continue;
        const float* gp = g + ((size_t)t2 * NN + n) * COUT;
        const float* wp = wt + (dt + 1) * COUT * COUT + o;
        #pragma unroll
        for (int oi = 0; oi < COUT; ++oi) s += gp[oi] * wp[oi * COUT];
    }
    out[(size_t)tn * COUT + o] = fmaxf(s, 0.f);
}

// ---------------- host ----------------

extern "C" void kernel_launch(void* const* d_in, const int* in_sizes, int n_in,
                              void* d_out, int out_size, void* d_ws, size_t ws_size,
                              hipStream_t stream) {
    const float* x     = (const float*)d_in[0];   // [B,T,N,C]
    const float* adj   = (const float*)d_in[1];   // [N,N]
    const float* att   = (const float*)d_in[2];   // [B,N,N]
    const float* theta = (const float*)d_in[3];   // [3,32,64]
    const float* cw    = (const float*)d_in[4];   // [64,64,1,3]
    const float* cb    = (const float*)d_in[5];   // [64]
    float* out = (float*)d_out;                   // [B,T,N,O]

    float* ws  = (float*)d_ws;
    float* vV   = ws;                               // 2048
    float* vY   = ws + 2048;                        // 2048
    float* scal = ws + 4096;                        // 16
    float* Lt   = ws + 8192;                        // 4M
    float* T2   = Lt + (size_t)NN * NN;             // 4M
    float* xt   = T2 + (size_t)NN * NN;             // B*N*T*C
    float* rhs  = xt + (size_t)BB * NN * JTOT;      // 3*N*J   (per-b, reused)
    float* g    = rhs + (size_t)KORD * NN * JTOT;   // T*N*O   (per-b, reused)

    // lambda_max via power iteration (dominant eig gap ~20x -> fast convergence)
    k_fill<<<8, 256, 0, stream>>>(vV, 0.02209708691f /*1/sqrt(2048)*/, NN);
    for (int it = 0; it < 12; ++it) {
        k_matvec<<<256, 256, 0, stream>>>(adj, vV, vY);
        k_reduce<<<1, 256, 0, stream>>>(vY, vV, scal, 0);
        k_scale<<<8, 256, 0, stream>>>(vY, scal, vV);
    }
    k_matvec<<<256, 256, 0, stream>>>(adj, vV, vY);
    k_reduce<<<1, 256, 0, stream>>>(vY, vV, scal, 1);   // scal[0] = lambda

    // L_tilde, T2 = 2*L@L - I
    k_ltilde<<<(NN * NN) / 256, 256, 0, stream>>>(adj, scal, Lt);
    k_gemm<false><<<dim3(NN / MT, NN / NTL), 256, 0, stream>>>(
        Lt, nullptr, Lt, T2, NN, 2.f, 1);

    // x -> [b][m][t][c]
    k_xt<<<(BB * NN * TT * CIN) / 256, 256, 0, stream>>>(x, xt);

    const int tnGrid = (TT * NN) / 4;
    for (int b = 0; b < BB; ++b) {
        const float* attb = att + (size_t)b * NN * NN;
        const float* xtb  = xt + (size_t)b * NN * JTOT;
        // k=0: identity-masked attention -> diagonal scale (no GEMM)
        k_rhs0<<<(NN * JTOT) / 256, 256, 0, stream>>>(attb, xtb, rhs);
        // k=1,2: fused (cheb ∘ att) @ xt
        k_gemm<true><<<dim3(NN / MT, JTOT / NTL), 256, 0, stream>>>(
            Lt, attb, xtb, rhs + (size_t)1 * NN * JTOT, JTOT, 1.f, 0);
        k_gemm<true><<<dim3(NN / MT, JTOT / NTL), 256, 0, stream>>>(
            T2, attb, xtb, rhs + (size_t)2 * NN * JTOT, JTOT, 1.f, 0);
        // theta + ReLU, then temporal conv + bias + ReLU
        k_theta<<<tnGrid, 256, 0, stream>>>(rhs, theta, g);
        k_conv<<<tnGrid, 256, 0, stream>>>(g, cw, cb,
                                           out + (size_t)b * TT * NN * COUT);
    }
}